// ReactionGCN_62732292326003
// MI455X (gfx1250) — compile-verified
//
#include <hip/hip_runtime.h>

#define F_IN 22
#define HDIM 64
#define EPSV 1e-5f

typedef __attribute__((ext_vector_type(2))) float v2f;
typedef __attribute__((ext_vector_type(8))) float v8f;

// ---------------------------------------------------------------- utilities
__global__ void zero_f32(float* __restrict__ p, int n) {
  int i = blockIdx.x * blockDim.x + threadIdx.x;
  if (i < n) p[i] = 0.f;
}

// deg accumulation over dst (self-loop +1 added in finalize)
__global__ void degree_accum(const int* __restrict__ dst, float* __restrict__ deg, int E) {
  int e = blockIdx.x * blockDim.x + threadIdx.x;
  if (e < E) atomicAdd(&deg[dst[e]], 1.0f);
}

__global__ void finalize_dinv(float* __restrict__ deg, int n) {
  int i = blockIdx.x * blockDim.x + threadIdx.x;
  if (i < n) deg[i] = rsqrtf(deg[i] + 1.0f);  // deg >= 1 due to self-loop
}

// ------------------------------------------------- WMMA fp32 GEMM (A @ W^T)
// C[nrows,64] = A[nrows,K] @ W[64,K]^T using V_WMMA_F32_16X16X4_F32.
// 256 threads = 8 waves; block tile = 32 rows x 64 cols; one 16x16 tile/wave.
template <int K, int KP>
__global__ __launch_bounds__(256) void wmma_gemm(const float* __restrict__ A,
                                                 const float* __restrict__ W,
                                                 float* __restrict__ C, int nrows) {
  __shared__ float sW[HDIM * KP];   // W zero-padded to KP
  __shared__ float sA[32 * KP];     // 32-row A tile, zero-padded
  const int tid = threadIdx.x;

  for (int idx = tid; idx < HDIM * KP; idx += 256) {
    int j = idx / KP, k = idx % KP;
    sW[idx] = (k < K) ? W[j * K + k] : 0.f;
  }
  const int row0 = blockIdx.x * 32;
  for (int idx = tid; idx < 32 * KP; idx += 256) {
    int r = idx / KP, k = idx % KP;
    int gr = row0 + r;
    sA[idx] = (gr < nrows && k < K) ? A[gr * K + k] : 0.f;
  }
  __syncthreads();

  const int wave = tid >> 5;       // 0..7
  const int lane = tid & 31;
  const int rt = wave >> 2;        // row tile 0..1
  const int ct = wave & 3;         // col tile 0..3
  const int hi = lane >> 4;        // lane half: 0 -> K={k0,k0+1}, 1 -> K={k0+2,k0+3}
  const int lr = lane & 15;

  const float* aRow = &sA[(rt * 16 + lr) * KP];  // A row M=lr of this tile
  const float* bRow = &sW[(ct * 16 + lr) * KP];  // B col N=lr  (B = W^T)

  v8f acc = {};
#pragma unroll
  for (int k0 = 0; k0 < KP; k0 += 4) {
    v2f a, b;
    a.x = aRow[k0 + 2 * hi];
    a.y = aRow[k0 + 2 * hi + 1];
    b.x = bRow[k0 + 2 * hi];
    b.y = bRow[k0 + 2 * hi + 1];
    acc = __builtin_amdgcn_wmma_f32_16x16x4_f32(false, a, false, b, (short)0, acc,
                                                false, false);
  }

  // C/D layout: VGPR v -> (M = v + 8*hi, N = lr)
  const int col = ct * 16 + lr;
  const int rbase = row0 + rt * 16 + hi * 8;
  if (row0 + 32 <= nrows) {
    // full tile (always the case when nrows % 32 == 0): branchless stores
#pragma unroll
    for (int v = 0; v < 8; ++v) C[(rbase + v) * HDIM + col] = acc[v];
  } else {
#pragma unroll
    for (int v = 0; v < 8; ++v) {
      int row = rbase + v;
      if (row < nrows) C[row * HDIM + col] = acc[v];
    }
  }
}

// -------------------------------------------- aggregation (self loop + bias)
__global__ void self_loop_init(float* __restrict__ out, const float* __restrict__ hlin,
                               const float* __restrict__ dinv, const float* __restrict__ bias,
                               int n) {
  int idx = blockIdx.x * blockDim.x + threadIdx.x;
  if (idx >= n * HDIM) return;
  int node = idx >> 6;
  int j = idx & 63;
  float di = dinv[node];
  out[idx] = hlin[idx] * di * di + bias[j];
}

// one wave (32 lanes) per edge; each lane scatters 2 features (float2 gather)
__global__ void edge_scatter(float* __restrict__ out, const float* __restrict__ hlin,
                             const float* __restrict__ dinv, const int* __restrict__ src,
                             const int* __restrict__ dst, int E) {
  int t = blockIdx.x * blockDim.x + threadIdx.x;
  int e = t >> 5;
  if (e >= E) return;
  int lane = t & 31;
  int s = src[e];
  int d = dst[e];
  float nm = dinv[s] * dinv[d];
  const float2* hp = (const float2*)(hlin + (size_t)s * HDIM);
  float2 v = hp[lane];
  atomicAdd(&out[(size_t)d * HDIM + 2 * lane], v.x * nm);
  atomicAdd(&out[(size_t)d * HDIM + 2 * lane + 1], v.y * nm);
}

// -------------------------------------------------------------- batch norm
__global__ void bn_stats(const float* __restrict__ h, float* __restrict__ stats, int n) {
  __shared__ float ssum[HDIM];
  __shared__ float ssq[HDIM];
  int tid = threadIdx.x;  // 256
  if (tid < HDIM) { ssum[tid] = 0.f; ssq[tid] = 0.f; }
  __syncthreads();
  int ch = tid & 63;
  float s = 0.f, q = 0.f;
  for (int r = blockIdx.x * 4 + (tid >> 6); r < n; r += gridDim.x * 4) {
    float v = h[(size_t)r * HDIM + ch];
    s += v;
    q += v * v;
  }
  atomicAdd(&ssum[ch], s);
  atomicAdd(&ssq[ch], q);
  __syncthreads();
  if (tid < HDIM) atomicAdd(&stats[tid], ssum[tid]);
  else if (tid < 2 * HDIM) atomicAdd(&stats[tid], ssq[tid - HDIM]);
}

__global__ void bn_relu(float* __restrict__ h, const float* __restrict__ stats,
                        const float* __restrict__ g, const float* __restrict__ be, int n) {
  int idx = blockIdx.x * blockDim.x + threadIdx.x;
  if (idx >= n * HDIM) return;
  int ch = idx & 63;
  float invN = 1.0f / (float)n;
  float mu = stats[ch] * invN;
  float var = stats[HDIM + ch] * invN - mu * mu;
  float v = h[idx];
  v = g[ch] * (v - mu) * rsqrtf(var + EPSV) + be[ch];
  h[idx] = fmaxf(v, 0.f);
}

// ------------------------------------------------------------------ pooling
__global__ void pool_kernel(float* __restrict__ gsum, float* __restrict__ gcnt,
                            const float* __restrict__ h, const int* __restrict__ batch, int n) {
  int t = blockIdx.x * blockDim.x + threadIdx.x;
  int node = t >> 5;
  if (node >= n) return;
  int lane = t & 31;
  int g = batch[node];
  const float2* hp = (const float2*)(h + (size_t)node * HDIM);
  float2 v = hp[lane];
  atomicAdd(&gsum[(size_t)g * HDIM + 2 * lane], v.x);
  atomicAdd(&gsum[(size_t)g * HDIM + 2 * lane + 1], v.y);
  if (lane == 0) atomicAdd(&gcnt[g], 1.0f);
}

// ----------------------------------------------------------------- MLP head
__global__ void head_kernel(float* __restrict__ out, const float* __restrict__ gsum,
                            const float* __restrict__ gcnt, const float* __restrict__ fcw1,
                            const float* __restrict__ fcb1, const float* __restrict__ fcw2,
                            const float* __restrict__ fcb2) {
  __shared__ float pooled[HDIM];
  __shared__ float z[HDIM / 2];
  int g = blockIdx.x;
  int tid = threadIdx.x;  // 64
  float inv = 1.0f / fmaxf(gcnt[g], 1.0f);
  pooled[tid] = gsum[(size_t)g * HDIM + tid] * inv;
  __syncthreads();
  if (tid < HDIM / 2) {
    float acc = fcb1[tid];
#pragma unroll
    for (int j = 0; j < HDIM; ++j) acc += pooled[j] * fcw1[tid * HDIM + j];
    z[tid] = fmaxf(acc, 0.f);
  }
  __syncthreads();
  if (tid == 0) {
    float o = fcb2[0];
#pragma unroll
    for (int i = 0; i < HDIM / 2; ++i) o += z[i] * fcw2[i];
    out[g] = o;
  }
}

// ------------------------------------------------------------------- launch
extern "C" void kernel_launch(void* const* d_in, const int* in_sizes, int n_in,
                              void* d_out, int out_size, void* d_ws, size_t ws_size,
                              hipStream_t stream) {
  (void)n_in; (void)ws_size;
  const float* x    = (const float*)d_in[0];
  const int*   ei   = (const int*)d_in[1];
  const int*   batch= (const int*)d_in[2];
  const float* W1   = (const float*)d_in[3];
  const float* b1   = (const float*)d_in[4];
  const float* g1   = (const float*)d_in[5];
  const float* be1  = (const float*)d_in[6];
  const float* W2   = (const float*)d_in[7];
  const float* b2   = (const float*)d_in[8];
  const float* g2   = (const float*)d_in[9];
  const float* be2  = (const float*)d_in[10];
  const float* fcw1 = (const float*)d_in[11];
  const float* fcb1 = (const float*)d_in[12];
  const float* fcw2 = (const float*)d_in[13];
  const float* fcb2 = (const float*)d_in[14];
  float* out = (float*)d_out;

  const int N = in_sizes[0] / F_IN;
  const int E = in_sizes[1] / 2;
  const int G = out_size;  // OUT == 1
  const int* src = ei;
  const int* dst = ei + E;

  // workspace carve-out
  char* base = (char*)d_ws;
  size_t off = 0;
  auto alloc = [&](size_t bytes) -> void* {
    off = (off + 255) & ~(size_t)255;
    void* p = base + off;
    off += bytes;
    return p;
  };
  float* dinv  = (float*)alloc(sizeof(float) * (size_t)N);
  float* hbuf1 = (float*)alloc(sizeof(float) * (size_t)N * HDIM);  // hlin
  float* hbuf2 = (float*)alloc(sizeof(float) * (size_t)N * HDIM);  // agg/activation
  float* stats = (float*)alloc(sizeof(float) * 2 * HDIM);
  float* gsum  = (float*)alloc(sizeof(float) * (size_t)G * HDIM);
  float* gcnt  = (float*)alloc(sizeof(float) * (size_t)G);

  const int NB  = (N + 255) / 256;
  const int NHB = (N * HDIM + 255) / 256;
  const int EB  = (int)(((size_t)E * 32 + 255) / 256);
  const int PB  = (int)(((size_t)N * 32 + 255) / 256);
  const int GB  = (N + 31) / 32;  // gemm row-tiles

  // degree -> dinv
  zero_f32<<<NB, 256, 0, stream>>>(dinv, N);
  zero_f32<<<(G * HDIM + 255) / 256, 256, 0, stream>>>(gsum, G * HDIM);
  zero_f32<<<(G + 255) / 256, 256, 0, stream>>>(gcnt, G);
  degree_accum<<<(E + 255) / 256, 256, 0, stream>>>(dst, dinv, E);
  finalize_dinv<<<NB, 256, 0, stream>>>(dinv, N);

  // ---- layer 1: x @ W1^T -> aggregate -> BN -> ReLU
  wmma_gemm<F_IN, 24><<<GB, 256, 0, stream>>>(x, W1, hbuf1, N);
  self_loop_init<<<NHB, 256, 0, stream>>>(hbuf2, hbuf1, dinv, b1, N);
  edge_scatter<<<EB, 256, 0, stream>>>(hbuf2, hbuf1, dinv, src, dst, E);
  zero_f32<<<1, 128, 0, stream>>>(stats, 2 * HDIM);
  bn_stats<<<512, 256, 0, stream>>>(hbuf2, stats, N);
  bn_relu<<<NHB, 256, 0, stream>>>(hbuf2, stats, g1, be1, N);

  // ---- layer 2: h1 @ W2^T -> aggregate -> BN -> ReLU
  wmma_gemm<HDIM, HDIM><<<GB, 256, 0, stream>>>(hbuf2, W2, hbuf1, N);
  self_loop_init<<<NHB, 256, 0, stream>>>(hbuf2, hbuf1, dinv, b2, N);
  edge_scatter<<<EB, 256, 0, stream>>>(hbuf2, hbuf1, dinv, src, dst, E);
  zero_f32<<<1, 128, 0, stream>>>(stats, 2 * HDIM);
  bn_stats<<<512, 256, 0, stream>>>(hbuf2, stats, N);
  bn_relu<<<NHB, 256, 0, stream>>>(hbuf2, stats, g2, be2, N);

  // ---- mean pool + MLP head
  pool_kernel<<<PB, 256, 0, stream>>>(gsum, gcnt, hbuf2, batch, N);
  head_kernel<<<G, 64, 0, stream>>>(out, gsum, gcnt, fcw1, fcb1, fcw2, fcb2);
}